// CurricularFacePenaltySMLoss_43808666419808
// MI455X (gfx1250) — compile-verified
//
#include <hip/hip_runtime.h>
#include <math.h>

#define B_ROWS 1024
#define C_COLS 85742
#define TPB    512
#define NBUF   4

// margin constants (M = 0.5, S = 64)
#define K_COSM   0.8775825618903728f    // cos(0.5)
#define K_SINM   0.4794255386042030f    // sin(0.5)
#define K_THRESH (-0.8775825618903728f) // cos(pi-0.5) = -cos(0.5)
#define K_MM     0.2397127693021015f    // sin(pi-0.5)*0.5
#define K_S      64.0f
#define K_TMOM   0.01f

// ---------------- CDNA5 async global->LDS helpers -----------------
#define AS1 __attribute__((address_space(1)))
#define AS3 __attribute__((address_space(3)))

typedef int v4i __attribute__((vector_size(16)));   // matches builtin prototype V4i

__device__ __forceinline__ void async_load16(const float4* g, float4* l) {
#if __has_builtin(__builtin_amdgcn_global_load_async_to_lds_b128)
  // (global src, lds dst, imm offset, imm cpol); cpol=1 -> TH_NT (stream, don't cache)
  __builtin_amdgcn_global_load_async_to_lds_b128((AS1 v4i*)(g), (AS3 v4i*)(l), 0, 1);
#else
  *l = *g;  // fallback: synchronous stage (still correct)
#endif
}

template <int N>
__device__ __forceinline__ void wait_async_le() {
#if __has_builtin(__builtin_amdgcn_s_wait_asynccnt)
  __builtin_amdgcn_s_wait_asynccnt(N);
#else
  if constexpr (N == 0)      asm volatile("s_wait_asynccnt 0x0" ::: "memory");
  else if constexpr (N == 1) asm volatile("s_wait_asynccnt 0x1" ::: "memory");
  else if constexpr (N == 2) asm volatile("s_wait_asynccnt 0x2" ::: "memory");
  else                       asm volatile("s_wait_asynccnt 0x3" ::: "memory");
#endif
}

// online logsumexp accumulate of z into (m, s): one exp per element
__device__ __forceinline__ void lse_acc(float z, float& m, float& s) {
  if (z <= m) {
    s += __expf(z - m);
  } else {
    s = s * __expf(m - z) + 1.0f;
    m = z;
  }
}

// ---------------- kernel: streaming per-row logsumexp (hot) -------
__global__ __launch_bounds__(TPB)
void cf_rowlse_kernel(const float* __restrict__ logits, const int* __restrict__ labels,
                      const float* __restrict__ ctm, const float* __restrict__ ftl,
                      const float* __restrict__ tval, float* __restrict__ rowloss) {
  const int r   = blockIdx.x;
  const int tid = threadIdx.x;
  const float* row = logits + (size_t)r * C_COLS;
  const int   lbl  = labels[r];
  const float c    = ctm[r];
  const float f    = ftl[r];
  const float t    = tval[0];

  // align body to 16B for b128 async stages
  const uintptr_t a = (uintptr_t)row;
  const int head  = (int)(((16u - (unsigned)(a & 15u)) & 15u) >> 2);
  const int nvec4 = (C_COLS - head) >> 2;
  const int tail  = (C_COLS - head) & 3;
  const float* vbase = row + head;
  const int niter = (nvec4 + TPB - 1) / TPB;

  __shared__ float4 stage[NBUF][TPB];  // 32 KB, 4-deep pipeline
  __shared__ float  red_m[TPB];
  __shared__ float  red_s[TPB];

  // prologue: issue chunks 0..NBUF-1 (clamped index -> all 32 lanes always issue)
#pragma unroll
  for (int p = 0; p < NBUF; ++p) {
    if (p < niter) {
      int vp = p * TPB + tid; if (vp >= nvec4) vp = nvec4 - 1;
      async_load16((const float4*)(vbase + 4 * (size_t)vp), &stage[p][tid]);
    }
  }

  float m = -3.402823e38f;
  float s = 0.0f;

  for (int i = 0; i < niter; ++i) {
    // chunk i complete when outstanding <= min(NBUF-1, chunks after i)
    const int rem = niter - 1 - i;
    if (rem >= 3)      wait_async_le<3>();
    else if (rem == 2) wait_async_le<2>();
    else if (rem == 1) wait_async_le<1>();
    else               wait_async_le<0>();
    asm volatile("" ::: "memory");

    const float4 v = stage[i & (NBUF - 1)][tid];
    const int my = i * TPB + tid;
    if (my < nvec4) {
      const int col0 = head + 4 * my;
#pragma unroll
      for (int k = 0; k < 4; ++k) {
        const float x = (&v.x)[k];
        float val = (x > c) ? x * (t + x) : x;
        if (col0 + k == lbl) val = f;      // scatter margin-adjusted target
        lse_acc(K_S * val, m, s);
      }
    }
    asm volatile("" ::: "memory");

    // refill the buffer just consumed (its LDS read completed: data was
    // consumed by VALU after the backend's dscnt wait, in program order)
    const int nx = i + NBUF;
    if (nx < niter) {
      int vn = nx * TPB + tid; if (vn >= nvec4) vn = nvec4 - 1;
      async_load16((const float4*)(vbase + 4 * (size_t)vn), &stage[i & (NBUF - 1)][tid]);
    }
  }

  // scalar head/tail handled by thread 0
  if (tid == 0) {
    for (int j = 0; j < head; ++j) {
      const float x = row[j];
      float val = (x > c) ? x * (t + x) : x;
      if (j == lbl) val = f;
      lse_acc(K_S * val, m, s);
    }
    const int tbase = head + 4 * nvec4;
    for (int j = 0; j < tail; ++j) {
      const float x = row[tbase + j];
      float val = (x > c) ? x * (t + x) : x;
      if (tbase + j == lbl) val = f;
      lse_acc(K_S * val, m, s);
    }
  }

  red_m[tid] = m;
  red_s[tid] = s;
  __syncthreads();
  for (int off = TPB / 2; off > 0; off >>= 1) {
    if (tid < off) {
      const float m1 = red_m[tid],       s1 = red_s[tid];
      const float m2 = red_m[tid + off], s2 = red_s[tid + off];
      const float M  = fmaxf(m1, m2);
      red_m[tid] = M;
      red_s[tid] = s1 * __expf(m1 - M) + s2 * __expf(m2 - M);
    }
    __syncthreads();
  }
  if (tid == 0) {
    const float lse = red_m[0] + logf(red_s[0]);
    rowloss[r] = lse - K_S * f;          // = -(numerator - log_denominator)
  }
}

// ---------------- kernel: per-row prep + t ------------------------
__global__ __launch_bounds__(B_ROWS)
void cf_prep_kernel(const float* __restrict__ logits, const int* __restrict__ labels,
                    float* __restrict__ ctm, float* __restrict__ ftl,
                    float* __restrict__ tval) {
  const int r = threadIdx.x;
  const int lbl = labels[r];
  const float tl = logits[(size_t)r * C_COLS + lbl];
  const float sn = sqrtf(fmaxf(0.0f, 1.0f - tl * tl));
  const float c  = tl * K_COSM - sn * K_SINM;
  ctm[r] = c;
  ftl[r] = (tl > K_THRESH) ? c : (tl - K_MM);

  __shared__ float red[B_ROWS];
  red[r] = tl;
  __syncthreads();
  for (int off = B_ROWS / 2; off > 0; off >>= 1) {
    if (r < off) red[r] += red[r + off];
    __syncthreads();
  }
  if (r == 0) tval[0] = K_TMOM * red[0] * (1.0f / (float)B_ROWS);
}

// ---------------- kernel: final mean ------------------------------
__global__ __launch_bounds__(B_ROWS)
void cf_finalize_kernel(const float* __restrict__ rowloss, float* __restrict__ out) {
  const int tr = threadIdx.x;
  __shared__ float red[B_ROWS];
  red[tr] = rowloss[tr];
  __syncthreads();
  for (int off = B_ROWS / 2; off > 0; off >>= 1) {
    if (tr < off) red[tr] += red[tr + off];
    __syncthreads();
  }
  if (tr == 0) out[0] = red[0] * (1.0f / (float)B_ROWS);
}

// ---------------- launcher ----------------------------------------
extern "C" void kernel_launch(void* const* d_in, const int* in_sizes, int n_in,
                              void* d_out, int out_size, void* d_ws, size_t ws_size,
                              hipStream_t stream) {
  (void)in_sizes; (void)n_in; (void)out_size; (void)ws_size;
  const float* logits = (const float*)d_in[0];
  const int*   labels = (const int*)d_in[1];   // jax default x64 off -> int32

  float* ws      = (float*)d_ws;
  float* ctm     = ws;                 // [B]
  float* ftl     = ws + B_ROWS;        // [B]
  float* rowloss = ws + 2 * B_ROWS;    // [B]
  float* tval    = ws + 3 * B_ROWS;    // [1]

  cf_prep_kernel<<<1, B_ROWS, 0, stream>>>(logits, labels, ctm, ftl, tval);
  cf_rowlse_kernel<<<B_ROWS, TPB, 0, stream>>>(logits, labels, ctm, ftl, tval, rowloss);
  cf_finalize_kernel<<<1, B_ROWS, 0, stream>>>(rowloss, (float*)d_out);
}